// NonParametricGraphOPD_15582141349978
// MI455X (gfx1250) — compile-verified
//
#include <hip/hip_runtime.h>

#define NTRAIN 2000
#define NGRAPH 2000
#define GFEAT  6
#define OPD2   65536   // 256*256
#define BATCH  2048

typedef __attribute__((ext_vector_type(2))) float v2f;
typedef __attribute__((ext_vector_type(8))) float v8f;

// ---------------------------------------------------------------------------
// Kernel A: one block per query. Phase 1 = argmin over obs_pos (first-index
// tie-break to match jnp.argmin). Phase 2 = inter[b,g] = graph_dic[idx,:]·alpha[:,g].
// Writes inter as [BATCH][8], cols 6..7 zeroed (K padding for WMMA).
// ---------------------------------------------------------------------------
__global__ __launch_bounds__(256) void nn_inter_kernel(
    const float* __restrict__ positions,   // [BATCH][2]
    const float* __restrict__ obs_pos,     // [NTRAIN][2]
    const float* __restrict__ graph_dic,   // [NTRAIN][NGRAPH]
    const float* __restrict__ alpha,       // [NGRAPH][GFEAT]
    float* __restrict__ inter)             // [BATCH][8]
{
    __shared__ float sdist[256];
    __shared__ int   sidx[256];
    __shared__ float sacc[GFEAT][256];

    const int b = blockIdx.x;
    const int t = threadIdx.x;

    const float px = positions[b * 2 + 0];
    const float py = positions[b * 2 + 1];

    // ---- Phase 1: 1-NN argmin ----
    float best = 3.4e38f;
    int   bi   = 0x7fffffff;
    for (int n = t; n < NTRAIN; n += 256) {
        float dx = px - obs_pos[n * 2 + 0];
        float dy = py - obs_pos[n * 2 + 1];
        float d  = dx * dx + dy * dy;
        if (d < best || (d == best && n < bi)) { best = d; bi = n; }
    }
    sdist[t] = best;
    sidx[t]  = bi;
    __syncthreads();
    #pragma unroll
    for (int s = 128; s > 0; s >>= 1) {
        if (t < s) {
            float d2 = sdist[t + s];
            int   i2 = sidx[t + s];
            if (d2 < sdist[t] || (d2 == sdist[t] && i2 < sidx[t])) {
                sdist[t] = d2;
                sidx[t]  = i2;
            }
        }
        __syncthreads();
    }
    const int idx = sidx[0];

    // ---- Phase 2: inter = graph_dic[idx,:] @ alpha ----
    const float* __restrict__ grow = graph_dic + (long)idx * NGRAPH;
    float acc0 = 0.f, acc1 = 0.f, acc2 = 0.f, acc3 = 0.f, acc4 = 0.f, acc5 = 0.f;
    for (int n = t; n < NGRAPH; n += 256) {
        float g = grow[n];
        const float* __restrict__ ar = alpha + n * GFEAT;
        acc0 += g * ar[0];
        acc1 += g * ar[1];
        acc2 += g * ar[2];
        acc3 += g * ar[3];
        acc4 += g * ar[4];
        acc5 += g * ar[5];
    }
    sacc[0][t] = acc0; sacc[1][t] = acc1; sacc[2][t] = acc2;
    sacc[3][t] = acc3; sacc[4][t] = acc4; sacc[5][t] = acc5;
    __syncthreads();
    #pragma unroll
    for (int s = 128; s > 0; s >>= 1) {
        if (t < s) {
            #pragma unroll
            for (int g = 0; g < GFEAT; ++g) sacc[g][t] += sacc[g][t + s];
        }
        __syncthreads();
    }
    if (t < 8) inter[b * 8 + t] = (t < GFEAT) ? sacc[t][0] : 0.f;
}

// ---------------------------------------------------------------------------
// Kernel B: out[b,p] = sum_g inter[b,g] * S[g,p] via V_WMMA_F32_16X16X4_F32.
// One 16(batch) x 16(pixel) tile per wave; K=6 padded to 8 -> two WMMA(K=4).
// VGPR layouts per CDNA5 ISA 7.12.2:
//   A (16x4 f32): lane l<16 holds (K0,K1) of row bm+l; lane>=16 holds (K2,K3).
//   B (4x16 f32): VGPR0 = row K0 (lanes 0-15) / K2 (lanes 16-31) at col p0+l;
//                 VGPR1 = row K1 / K3.
//   D (16x16 f32): VGPR r -> row r (lanes 0-15) / r+8 (lanes 16-31), col p0+l.
// ---------------------------------------------------------------------------
__global__ __launch_bounds__(256) void expand_wmma_kernel(
    const float* __restrict__ inter,  // [BATCH][8]
    const float* __restrict__ S,      // [GFEAT][OPD2]
    float* __restrict__ out)          // [BATCH][OPD2]
{
    const int wave = threadIdx.x >> 5;
    const int lane = threadIdx.x & 31;
    const unsigned tile = blockIdx.x * 8u + wave;   // 0 .. 524287
    const int ptile = (int)(tile & 4095u);          // 4096 tiles along pixels
    const int btile = (int)(tile >> 12);            // 128 tiles along batch
    const int p0 = ptile * 16;
    const int bm = btile * 16;

    const int half = lane >> 4;   // 0: lanes 0-15, 1: lanes 16-31
    const int l    = lane & 15;

    // A operands (fp32, full precision)
    const float* __restrict__ arow = inter + (bm + l) * 8;
    v2f a0, a1;
    a0.x = arow[half * 2 + 0];        // K = 0/2
    a0.y = arow[half * 2 + 1];        // K = 1/3
    a1.x = arow[4 + half * 2 + 0];    // K = 4/6  (6 is zero-padded)
    a1.y = arow[4 + half * 2 + 1];    // K = 5/7  (7 is zero-padded)

    // B operands. For the second WMMA, lanes>=16 would need rows 6,7 which do
    // not exist; their A values are 0, so any finite in-bounds value works —
    // re-read rows 4,5 (0 * finite = 0).
    v2f b0, b1;
    b0.x = S[(half * 2 + 0) * OPD2 + p0 + l];
    b0.y = S[(half * 2 + 1) * OPD2 + p0 + l];
    b1.x = S[4 * OPD2 + p0 + l];      // K=4 (half 0) / don't-care (half 1)
    b1.y = S[5 * OPD2 + p0 + l];      // K=5 (half 0) / don't-care (half 1)

    v8f c = {0.f, 0.f, 0.f, 0.f, 0.f, 0.f, 0.f, 0.f};
    c = __builtin_amdgcn_wmma_f32_16x16x4_f32(false, a0, false, b0,
                                              (short)0, c, false, false);
    c = __builtin_amdgcn_wmma_f32_16x16x4_f32(false, a1, false, b1,
                                              (short)0, c, false, false);

    // Store: VGPR r -> out[bm + r + 8*half][p0 + l]; NT hint to stream past L2.
    float* __restrict__ obase = out + (long)(bm + half * 8) * OPD2 + p0 + l;
    #pragma unroll
    for (int r = 0; r < 8; ++r)
        __builtin_nontemporal_store(c[r], obase + (long)r * OPD2);
}

extern "C" void kernel_launch(void* const* d_in, const int* in_sizes, int n_in,
                              void* d_out, int out_size, void* d_ws, size_t ws_size,
                              hipStream_t stream) {
    const float* positions = (const float*)d_in[0];  // [2048,2]
    const float* obs_pos   = (const float*)d_in[1];  // [2000,2]
    const float* graph_dic = (const float*)d_in[2];  // [2000,2000]
    const float* alpha     = (const float*)d_in[3];  // [2000,6]
    const float* S_graph   = (const float*)d_in[4];  // [6,256,256]
    float* out   = (float*)d_out;                    // [2048,256,256]
    float* inter = (float*)d_ws;                     // [2048,8] = 64 KB scratch

    nn_inter_kernel<<<BATCH, 256, 0, stream>>>(positions, obs_pos, graph_dic,
                                               alpha, inter);

    // tiles: (BATCH/16) * (OPD2/16) = 128 * 4096 = 524288; 8 waves per block
    const int nblocks = (BATCH / 16) * (OPD2 / 16) / 8;  // 65536
    expand_wmma_kernel<<<nblocks, 256, 0, stream>>>(inter, S_graph, out);
}